// FlowLeniaParams_41240275976335
// MI455X (gfx1250) — compile-verified
//
#include <hip/hip_runtime.h>
#include <math.h>

// ---------------------------------------------------------------------------
// Flow Lenia step for MI455X (gfx1250, wave32).
// FFTs run as four-step 256 = 16x16 transforms on the matrix pipes via
// V_WMMA_F32_16X16X4_F32 (full fp32 WMMA). One wave32 owns one 256-pt FFT.
// DFT-matrix A-fragments are generated once per wave and reused by both
// matmul stages (they are identical), halving the sin/cos VALU load.
// ---------------------------------------------------------------------------

typedef __attribute__((ext_vector_type(2))) float v2f;
typedef __attribute__((ext_vector_type(8))) float v8f;

#define NPIX   65536          // 256*256
#define MA     4.35f          // DD - SIGMA
#define SIGMA_ 0.65f
#define DT_    0.2f

// 8 args: (neg_a, A, neg_b, B, c_mod, C, reuse_a, reuse_b)
#define WMMA_F32(a, b, c) \
    __builtin_amdgcn_wmma_f32_16x16x4_f32(false, (a), false, (b), (short)0, (c), false, false)

// ---------------------------------------------------------------------------
// 256-point complex FFT, one wave, data in per-wave LDS scratch (256 re + 256 im).
// sign = -1 forward, +1 inverse (no 1/N scaling applied here).
// Four-step: Y = D16*X ; Z = Y .* W256^(n1*m2) ; Out[n2][n1] = D16 * Z^T
// Output written back to LDS in natural order.
// ---------------------------------------------------------------------------
__device__ __forceinline__ void fft256_wave(float* re, float* im, int lane, float sign)
{
    const int col  = lane & 15;   // M-row for A frags, N-column for B/C/D frags
    const int half = lane >> 4;
    const float W16A  = 0.39269908169872414f;   // 2*pi/16
    const float W256A = 0.024543692606170259f;  // 2*pi/256

    // ---- DFT-matrix A-fragments (shared by both stages): D[row][k]=e^{sign*i*2pi*row*k/16}
    v2f dr[4], di[4], dni[4];
#pragma unroll
    for (int kc = 0; kc < 4; ++kc) {
        const int k0 = 4 * kc + 2 * half;   // K indices carried by this lane
        const int k1 = k0 + 1;
        float s0, c0, s1, c1;
        __sincosf(W16A * (float)((col * k0) & 15), &s0, &c0);
        __sincosf(W16A * (float)((col * k1) & 15), &s1, &c1);
        dr[kc].x  = c0;          dr[kc].y  = c1;
        di[kc].x  = sign * s0;   di[kc].y  = sign * s1;
        dni[kc].x = -di[kc].x;   dni[kc].y = -di[kc].y;
    }

    // ---- Stage A: Y[n1][m2] = sum_m1 D[n1][m1] * X[m1][m2], X[m1][m2]=lds[16*m1+m2]
    v8f yr = {}, yi = {};
#pragma unroll
    for (int kc = 0; kc < 4; ++kc) {
        const int k0 = 4 * kc + 2 * half;
        const int k1 = k0 + 1;
        v2f bxr, bxi;                       // B frag: rows k0,k1 of X, column `col`
        bxr.x = re[16 * k0 + col]; bxr.y = re[16 * k1 + col];
        bxi.x = im[16 * k0 + col]; bxi.y = im[16 * k1 + col];
        yr = WMMA_F32(dr[kc],  bxr, yr);    // Re += Dr*Xr
        yr = WMMA_F32(dni[kc], bxi, yr);    // Re -= Di*Xi
        yi = WMMA_F32(dr[kc],  bxi, yi);    // Im += Dr*Xi
        yi = WMMA_F32(di[kc],  bxr, yi);    // Im += Di*Xr
    }

    // ---- Twiddle + store Z back to LDS (in place; same-wave DS ops are in order)
#pragma unroll
    for (int g = 0; g < 8; ++g) {
        const int n1 = g + 8 * half;        // C/D layout: VGPR g -> row g / g+8
        const int m2 = col;
        float sn, cs;
        __sincosf(W256A * (float)((n1 * m2) & 255), &sn, &cs);
        sn *= sign;
        const float zr = yr[g] * cs - yi[g] * sn;
        const float zi = yr[g] * sn + yi[g] * cs;
        re[16 * n1 + m2] = zr;
        im[16 * n1 + m2] = zi;
    }

    // ---- Stage B: Out[n2][n1] = sum_m2 D[n2][m2] * Z[n1][m2]  (B = Z^T from LDS)
    v8f orr = {}, oii = {};
#pragma unroll
    for (int kc = 0; kc < 4; ++kc) {
        const int k0 = 4 * kc + 2 * half;
        const int k1 = k0 + 1;
        v2f bzr, bzi;                       // Z^T[k][col] = Z[col][k]
        bzr.x = re[16 * col + k0]; bzr.y = re[16 * col + k1];
        bzi.x = im[16 * col + k0]; bzi.y = im[16 * col + k1];
        orr = WMMA_F32(dr[kc],  bzr, orr);
        orr = WMMA_F32(dni[kc], bzi, orr);
        oii = WMMA_F32(dr[kc],  bzi, oii);
        oii = WMMA_F32(di[kc],  bzr, oii);
    }

    // ---- Write back natural order: Out[n2][n1] = X[n1 + 16*n2]
#pragma unroll
    for (int g = 0; g < 8; ++g) {
        const int n2 = g + 8 * half;
        const int n  = col + 16 * n2;
        re[n] = orr[g];
        im[n] = oii[g];
    }
}

// ---------------------------------------------------------------------------
// Row FFTs of A: input A (X,Y,C) interleaved, output planar complex planes [c].
// 4 waves / block, one FFT per wave.
// ---------------------------------------------------------------------------
__global__ void k_fft_rows_A(const float* __restrict__ A,
                             float* __restrict__ bre, float* __restrict__ bim)
{
    __shared__ float lre[4 * 256];
    __shared__ float lim[4 * 256];
    const int w = threadIdx.x >> 5, lane = threadIdx.x & 31;
    const int gw = blockIdx.x * 4 + w;    // 0..767
    const int c = gw >> 8, x = gw & 255;
    float* re = lre + w * 256;
    float* im = lim + w * 256;
#pragma unroll
    for (int j = 0; j < 8; ++j) {
        const int m = lane + 32 * j;
        re[m] = A[(x * 256 + m) * 3 + c];
        im[m] = 0.f;
    }
    fft256_wave(re, im, lane, -1.f);
#pragma unroll
    for (int j = 0; j < 8; ++j) {
        const int m = lane + 32 * j;
        bre[c * NPIX + x * 256 + m] = re[m];
        bim[c * NPIX + x * 256 + m] = im[m];
    }
}

// ---------------------------------------------------------------------------
// Column FFTs over planar complex planes (used forward c=3 and inverse k=15).
// ---------------------------------------------------------------------------
__global__ void k_fft_cols(const float* __restrict__ inre, const float* __restrict__ inim,
                           float* __restrict__ outre, float* __restrict__ outim, float sign)
{
    __shared__ float lre[4 * 256];
    __shared__ float lim[4 * 256];
    const int w = threadIdx.x >> 5, lane = threadIdx.x & 31;
    const int gw = blockIdx.x * 4 + w;
    const int p = gw >> 8, y = gw & 255;
    float* re = lre + w * 256;
    float* im = lim + w * 256;
#pragma unroll
    for (int j = 0; j < 8; ++j) {
        const int m = lane + 32 * j;
        re[m] = inre[p * NPIX + m * 256 + y];
        im[m] = inim[p * NPIX + m * 256 + y];
    }
    fft256_wave(re, im, lane, sign);
#pragma unroll
    for (int j = 0; j < 8; ++j) {
        const int m = lane + 32 * j;
        outre[p * NPIX + m * 256 + y] = re[m];
        outim[p * NPIX + m * 256 + y] = im[m];
    }
}

// ---------------------------------------------------------------------------
// Final inverse row FFTs; keep real part scaled by 1/65536 (ifft2 norm).
// ---------------------------------------------------------------------------
__global__ void k_fft_rows_realout(const float* __restrict__ inre, const float* __restrict__ inim,
                                   float* __restrict__ U)
{
    __shared__ float lre[4 * 256];
    __shared__ float lim[4 * 256];
    const int w = threadIdx.x >> 5, lane = threadIdx.x & 31;
    const int gw = blockIdx.x * 4 + w;
    const int p = gw >> 8, x = gw & 255;
    float* re = lre + w * 256;
    float* im = lim + w * 256;
#pragma unroll
    for (int j = 0; j < 8; ++j) {
        const int m = lane + 32 * j;
        re[m] = inre[p * NPIX + x * 256 + m];
        im[m] = inim[p * NPIX + x * 256 + m];
    }
    fft256_wave(re, im, lane, 1.f);
#pragma unroll
    for (int j = 0; j < 8; ++j) {
        const int m = lane + 32 * j;
        U[p * NPIX + x * 256 + m] = re[m] * (1.f / 65536.f);
    }
}

// ---------------------------------------------------------------------------
// Spectral multiply: G[k] = fK[:, :, k] * fA[k % 3]
// ---------------------------------------------------------------------------
__global__ void k_combine(const float* __restrict__ fAre, const float* __restrict__ fAim,
                          const float* __restrict__ fKre, const float* __restrict__ fKim,
                          float* __restrict__ Gre, float* __restrict__ Gim)
{
    const int idx = blockIdx.x * 256 + threadIdx.x;   // spatial index x*256+y
#pragma unroll
    for (int k = 0; k < 15; ++k) {
        const int c = k % 3;
        const float ar = fAre[c * NPIX + idx], ai = fAim[c * NPIX + idx];
        const float kr = fKre[idx * 15 + k],   ki = fKim[idx * 15 + k];
        Gre[k * NPIX + idx] = kr * ar - ki * ai;
        Gim[k * NPIX + idx] = kr * ai + ki * ar;
    }
}

// ---------------------------------------------------------------------------
// Growth + per-channel kernel-group reduction: Uc[c] = sum_{k%3==c} growth(U_k)*P_k
// ---------------------------------------------------------------------------
__global__ void k_growth(const float* __restrict__ U, const float* __restrict__ P,
                         const float* __restrict__ m, const float* __restrict__ s,
                         float* __restrict__ Uc)
{
    const int idx = blockIdx.x * 256 + threadIdx.x;
    float acc[3] = {0.f, 0.f, 0.f};
#pragma unroll
    for (int k = 0; k < 15; ++k) {
        const float u = U[k * NPIX + idx];
        const float d = u - m[k];
        const float sk = s[k];
        const float g = 2.f * __expf(-(d * d) / (2.f * sk * sk)) - 1.f;
        acc[k % 3] += g * P[idx * 15 + k];
    }
    Uc[0 * NPIX + idx] = acc[0];
    Uc[1 * NPIX + idx] = acc[1];
    Uc[2 * NPIX + idx] = acc[2];
}

// ---------------------------------------------------------------------------
// Sobel flow (zero-padded 'SAME'), alpha mixing, clip, -> target centers mu.
// mus planes: (d*3 + c), d=0 row-dim, d=1 col-dim.
// ---------------------------------------------------------------------------
__global__ void k_flow(const float* __restrict__ Uc, const float* __restrict__ A,
                       float* __restrict__ mus)
{
    const int idx = blockIdx.x * 256 + threadIdx.x;
    const int x = idx >> 8, y = idx & 255;

    float un[3][3][3];   // [c][a+1][b+1]
    float an[3][3];
#pragma unroll
    for (int a = -1; a <= 1; ++a) {
#pragma unroll
        for (int b = -1; b <= 1; ++b) {
            const int xi = x + a, yi = y + b;
            const bool in = (xi >= 0) & (xi < 256) & (yi >= 0) & (yi < 256);
            const int ii = in ? (xi * 256 + yi) : 0;
            float asum = 0.f;
#pragma unroll
            for (int c = 0; c < 3; ++c) {
                un[c][a + 1][b + 1] = in ? Uc[c * NPIX + ii] : 0.f;
                asum += in ? A[ii * 3 + c] : 0.f;
            }
            an[a + 1][b + 1] = asum;
        }
    }
    // sobel(channel-summed A)
    const float cg0 = (an[2][0] + 2.f * an[2][1] + an[2][2]) - (an[0][0] + 2.f * an[0][1] + an[0][2]);
    const float cg1 = (an[0][2] + 2.f * an[1][2] + an[2][2]) - (an[0][0] + 2.f * an[1][0] + an[2][0]);

#pragma unroll
    for (int c = 0; c < 3; ++c) {
        const float f0 = (un[c][2][0] + 2.f * un[c][2][1] + un[c][2][2])
                       - (un[c][0][0] + 2.f * un[c][0][1] + un[c][0][2]);
        const float f1 = (un[c][0][2] + 2.f * un[c][1][2] + un[c][2][2])
                       - (un[c][0][0] + 2.f * un[c][1][0] + un[c][2][0]);
        float al = A[idx * 3 + c] * 0.5f;
        al = fminf(al * al, 1.f);
        float F0 = fminf(fmaxf(f0 * (1.f - al) - cg0 * al, -MA), MA);
        float F1 = fminf(fmaxf(f1 * (1.f - al) - cg1 * al, -MA), MA);
        F0 = fminf(fmaxf(DT_ * F0, -MA), MA);
        F1 = fminf(fmaxf(DT_ * F1, -MA), MA);
        const float mu0 = fminf(fmaxf((float)x + 0.5f + F0, SIGMA_), 256.f - SIGMA_);
        const float mu1 = fminf(fmaxf((float)y + 0.5f + F1, SIGMA_), 256.f - SIGMA_);
        mus[(0 * 3 + c) * NPIX + idx] = mu0;
        mus[(1 * 3 + c) * NPIX + idx] = mu1;
    }
}

// ---------------------------------------------------------------------------
// Reintegration: 16x16 tile/block, 26x26 halo tile in LDS (24 floats/cell =
// 63.4 KB), 121-tap toroidal gather with softmax mix.
// ---------------------------------------------------------------------------
#define SRC 26
__global__ void k_reint(const float* __restrict__ A, const float* __restrict__ P,
                        const float* __restrict__ mus,
                        float* __restrict__ outA, float* __restrict__ outP)
{
    __shared__ float sm[SRC * SRC * 24];
    const int tid = threadIdx.x;
    const int tileX = blockIdx.x >> 4, tileY = blockIdx.x & 15;

    for (int t = tid; t < SRC * SRC; t += 256) {
        const int sx = t / SRC, sy = t % SRC;
        const int gx = (tileX * 16 + sx - 5 + 256) & 255;   // torus wrap (jnp.roll)
        const int gy = (tileY * 16 + sy - 5 + 256) & 255;
        const int g = gx * 256 + gy;
        float* d = sm + t * 24;
#pragma unroll
        for (int c = 0; c < 3; ++c)  d[c]     = A[g * 3 + c];
#pragma unroll
        for (int q = 0; q < 6; ++q)  d[3 + q] = mus[q * NPIX + g];
#pragma unroll
        for (int k = 0; k < 15; ++k) d[9 + k] = P[g * 15 + k];
    }
    __syncthreads();

    const int lx = tid >> 4, ly = tid & 15;
    const int gx0 = tileX * 16 + lx, gy0 = tileY * 16 + ly;
    const float px = (float)gx0 + 0.5f, py = (float)gy0 + 0.5f;
    const float inv4s2 = 1.f / (4.f * SIGMA_ * SIGMA_);   // 1/1.69

    float nA0 = 0.f, nA1 = 0.f, nA2 = 0.f, accW = 0.f;
    float accP[15];
#pragma unroll
    for (int k = 0; k < 15; ++k) accP[k] = 0.f;

    for (int ox = 0; ox < 11; ++ox) {
        for (int oy = 0; oy < 11; ++oy) {
            const float* d = sm + (((lx + ox) * SRC) + (ly + oy)) * 24;
            // area_c = prod_d clamp(0.5 + sigma - |pos_d - mu_d|, 0, 1) / (4 sigma^2)
            float m0, m1, m2;
            {
                const float a0 = fminf(fmaxf(1.15f - fabsf(px - d[3 + 0]), 0.f), 1.f);
                const float b0 = fminf(fmaxf(1.15f - fabsf(py - d[6 + 0]), 0.f), 1.f);
                m0 = d[0] * (a0 * b0 * inv4s2);
                const float a1 = fminf(fmaxf(1.15f - fabsf(px - d[3 + 1]), 0.f), 1.f);
                const float b1 = fminf(fmaxf(1.15f - fabsf(py - d[6 + 1]), 0.f), 1.f);
                m1 = d[1] * (a1 * b1 * inv4s2);
                const float a2 = fminf(fmaxf(1.15f - fabsf(px - d[3 + 2]), 0.f), 1.f);
                const float b2 = fminf(fmaxf(1.15f - fabsf(py - d[6 + 2]), 0.f), 1.f);
                m2 = d[2] * (a2 * b2 * inv4s2);
            }
            const float e = __expf(m0 + m1 + m2) - 1.f;   // softmax mix weight
            nA0 += m0; nA1 += m1; nA2 += m2;
            accW += e;
#pragma unroll
            for (int k = 0; k < 15; ++k) accP[k] += d[9 + k] * e;
        }
    }

    const int g = gx0 * 256 + gy0;
    outA[g * 3 + 0] = nA0;
    outA[g * 3 + 1] = nA1;
    outA[g * 3 + 2] = nA2;
    const float inv = 1.f / (accW + 1e-10f);
#pragma unroll
    for (int k = 0; k < 15; ++k) outP[g * 15 + k] = accP[k] * inv;
}

// ---------------------------------------------------------------------------
extern "C" void kernel_launch(void* const* d_in, const int* in_sizes, int n_in,
                              void* d_out, int out_size, void* d_ws, size_t ws_size,
                              hipStream_t stream)
{
    (void)in_sizes; (void)n_in; (void)out_size; (void)ws_size;
    const float* A   = (const float*)d_in[0];
    const float* P   = (const float*)d_in[1];
    const float* fKr = (const float*)d_in[2];
    const float* fKi = (const float*)d_in[3];
    const float* m   = (const float*)d_in[4];
    const float* s   = (const float*)d_in[5];
    float* outA = (float*)d_out;
    float* outP = outA + 3 * NPIX;

    // Workspace: two planar complex ping-pong buffers (15 planes each) = 15.7 MB
    float* B0re = (float*)d_ws;
    float* B0im = B0re + 15 * NPIX;
    float* B1re = B0im + 15 * NPIX;
    float* B1im = B1re + 15 * NPIX;
    float* Uc   = B1re;               // reuse planes 0..2 after spectral phase
    float* MUS  = B1re + 3 * NPIX;    // planes 3..8

    k_fft_rows_A      <<<192, 128, 0, stream>>>(A, B0re, B0im);                    // fft rows (3 ch)
    k_fft_cols        <<<192, 128, 0, stream>>>(B0re, B0im, B1re, B1im, -1.f);     // fft cols -> fA
    k_combine         <<<256, 256, 0, stream>>>(B1re, B1im, fKr, fKi, B0re, B0im); // fK * fA[c0]
    k_fft_cols        <<<960, 128, 0, stream>>>(B0re, B0im, B1re, B1im, 1.f);      // ifft cols (15)
    k_fft_rows_realout<<<960, 128, 0, stream>>>(B1re, B1im, B0re);                 // ifft rows -> U
    k_growth          <<<256, 256, 0, stream>>>(B0re, P, m, s, Uc);                // growth * P, group sum
    k_flow            <<<256, 256, 0, stream>>>(Uc, A, MUS);                       // sobel -> mu
    k_reint           <<<256, 256, 0, stream>>>(A, P, MUS, outA, outP);            // 121-tap reintegration
}